// TreeProductQuantizer_68118181314716
// MI455X (gfx1250) — compile-verified
//
#include <hip/hip_runtime.h>
#include <hip/hip_bf16.h>

// Problem constants (match reference)
#define BB     16
#define TT     4096
#define DD     512
#define GG     8
#define DEPTH_ 8
#define GD     64
#define MTOK   (BB * TT)          // 65536 tokens
#define TILE   16                 // tokens per workgroup
#define NTILES (MTOK / TILE)      // 4096
#define XQ_ELEMS ((size_t)MTOK * DD)   // 33554432
#define LOSS_OFF XQ_ELEMS
#define IDX_OFF  (XQ_ELEMS + 1)

typedef __attribute__((ext_vector_type(2))) float v2f;
typedef __attribute__((ext_vector_type(4))) float v4f;
typedef __attribute__((ext_vector_type(8))) float v8f;

// ---------------------------------------------------------------------------
// Setup: Gram[g][e][f] = dot(levels[g,e/2,e&1,:], levels[g,f/2,f&1,:])  (e=k*2+c)
// Also zeroes the loss accumulator slot in d_out.
// ---------------------------------------------------------------------------
__global__ __launch_bounds__(256) void tpq_setup(const float* __restrict__ levels,
                                                 float* __restrict__ gram,
                                                 float* __restrict__ out)
{
    const int g   = blockIdx.x;        // 8 blocks
    const int tid = threadIdx.x;       // 256 = 16x16 (e,f) pairs
    const int e   = tid >> 4;
    const int f   = tid & 15;
    const float* ve = levels + (size_t)(g * 16 + e) * GD;
    const float* vf = levels + (size_t)(g * 16 + f) * GD;
    float s = 0.f;
#pragma unroll
    for (int d = 0; d < GD; ++d) s = fmaf(ve[d], vf[d], s);
    gram[(g * 16 + e) * 16 + f] = s;
    if (g == 0 && tid == 0) out[LOSS_OFF] = 0.f;
}

// ---------------------------------------------------------------------------
// Main kernel: one workgroup = 16 tokens; wave w handles group w.
// Phase 1: P = X_tile x V^T via 16 chained v_wmma_f32_16x16x4_f32 (K=64)
// Phase 2: greedy tree decisions via Gram recurrence (scalar, LDS tables)
// Phase 3: X_q = OneHot x V via 16 WMMAs; STE output + loss accumulation
// ---------------------------------------------------------------------------
__global__ __launch_bounds__(256) void tpq_main(const float* __restrict__ x,
                                                const float* __restrict__ levels,
                                                const float* __restrict__ gram,
                                                float* __restrict__ out)
{
    constexpr int XPITCH = 516;  // pad: 516*4B mod 256B = 16 -> conflict-free column gathers
    __shared__ __align__(16) float x_sh[TILE * XPITCH];   // ~33 KB
    __shared__ float P_sh[GG][TILE][17];                  // dots, padded
    __shared__ float G_sh[GG][16][16];                    // Gram tables
    __shared__ float red_sh[256];

    const int tile = blockIdx.x;
    const int tid  = threadIdx.x;
    const int lane = tid & 31;
    const int g    = tid >> 5;      // wave id == group id
    const int t16  = lane & 15;
    const int h    = lane >> 4;     // lane half (WMMA K split)

    // ---- Stage contiguous 32KB x tile into LDS (coalesced b128), Gram into LDS
    const float* xt = x + (size_t)tile * (TILE * DD);
#pragma unroll
    for (int i = 0; i < 8; ++i) {
        const int e4   = tid + i * 256;      // float4 index 0..2047
        const int elem = e4 * 4;
        const int row  = elem >> 9;          // /512
        const int col  = elem & 511;
        v4f v = *(const v4f*)(xt + elem);
        *(v4f*)(&x_sh[row * XPITCH + col]) = v;
    }
#pragma unroll
    for (int i = 0; i < 8; ++i) {
        const int k = tid + i * 256;
        ((float*)G_sh)[k] = gram[k];
    }
    // Stream-prefetch next tile (global_prefetch_b8): 256 lanes x 128B = 32KB
    if (tile + 1 < NTILES)
        __builtin_prefetch(x + (size_t)(tile + 1) * (TILE * DD) + tid * 32, 0, 3);
    __syncthreads();

    const float* lv = levels + (size_t)g * 16 * GD;   // group codebook (16 x 64)

    // ---- Phase 1: P(16 tok x 16 codewords) = X(16x64) * V^T(64x16), K in 16 steps of 4
    v2f bfrag[16];
#pragma unroll
    for (int s = 0; s < 16; ++s)   // B: rows K=dims, cols N=codewords; lane -> N=t16, K=4s+2h{,+1}
        bfrag[s] = *(const v2f*)(lv + t16 * GD + 4 * s + 2 * h);

    v8f c = {};
#pragma unroll
    for (int s = 0; s < 16; ++s) {
        // A: rows M=tokens (lane -> M=t16), cols K=dims 4s+2h{,+1}
        v2f a = *(const v2f*)(&x_sh[t16 * XPITCH + g * GD + 4 * s + 2 * h]);
        c = __builtin_amdgcn_wmma_f32_16x16x4_f32(false, a, false, bfrag[s],
                                                  (short)0, c, false, false);
    }
    // C layout: VGPR j -> token j+8h, col t16
#pragma unroll
    for (int j = 0; j < 8; ++j)
        P_sh[g][j + 8 * h][t16] = c[j];

    // ---- Phase 2: greedy residual-tree decisions via Gram recurrence
    // bit_k = (p1 - |v1|^2/2 > p0 - |v0|^2/2), then p[m>2k+1] -= Gram[chosen][m]
    float p[16];
#pragma unroll
    for (int m = 0; m < 16; ++m) p[m] = P_sh[g][t16][m];
    int idxv = 0;
#pragma unroll
    for (int k = 0; k < 8; ++k) {
        const float s0 = p[2 * k]     - 0.5f * G_sh[g][2 * k][2 * k];
        const float s1 = p[2 * k + 1] - 0.5f * G_sh[g][2 * k + 1][2 * k + 1];
        const int bit = (s1 > s0) ? 1 : 0;      // matches reference d1 < d0
        const int e   = 2 * k + bit;
        idxv |= bit << k;
#pragma unroll
        for (int m = 2 * k + 2; m < 16; ++m) p[m] -= G_sh[g][e][m];
    }
    if (lane < 16)   // lanes 16..31 computed the same token redundantly
        ((int*)out)[IDX_OFF + (size_t)(tile * TILE + t16) * GG + g] = idxv;

    // ---- Phase 3: X_q = OneHot(16x16) x V(16x64) via WMMA; STE store + loss
    float acc = 0.f;
    float* xq = out + (size_t)tile * TILE * DD + g * GD;
#pragma unroll
    for (int nb = 0; nb < 4; ++nb) {           // 4 blocks of 16 dims
        v8f cr = {};
#pragma unroll
        for (int s = 0; s < 4; ++s) {          // K = codewords, 4 per step
            const int k   = 2 * s + h;         // level index for this lane's K pair
            const int bit = (idxv >> k) & 1;
            v2f a;                             // one-hot: e=2k (c=0), e=2k+1 (c=1)
            a.x = bit ? 0.f : 1.f;
            a.y = bit ? 1.f : 0.f;
            const int e0 = 4 * s + 2 * h;
            v2f b;                             // B: rows K=codewords, cols N=dims
            b.x = lv[(e0    ) * GD + nb * 16 + t16];
            b.y = lv[(e0 + 1) * GD + nb * 16 + t16];
            cr = __builtin_amdgcn_wmma_f32_16x16x4_f32(false, a, false, b,
                                                       (short)0, cr, false, false);
        }
#pragma unroll
        for (int j = 0; j < 8; ++j) {
            const int tok = j + 8 * h;
            const float xv   = x_sh[tok * XPITCH + g * GD + nb * 16 + t16];
            const float diff = cr[j] - xv;             // x_q - x
            acc = fmaf(diff, diff, acc);
            xq[(size_t)tok * DD + nb * 16 + t16] = xv + diff;   // STE: x + sg(x_q - x)
        }
    }

    // ---- loss reduction: loss = 2 * sum(diff^2) / (B*T*GD), scaled per-contribution
    constexpr float LOSS_SCALE = 2.0f / (float)((size_t)MTOK * GD);
    red_sh[tid] = acc * LOSS_SCALE;
    __syncthreads();
#pragma unroll
    for (int off = 128; off > 0; off >>= 1) {
        if (tid < off) red_sh[tid] += red_sh[tid + off];
        __syncthreads();
    }
    if (tid == 0) atomicAdd(out + LOSS_OFF, red_sh[0]);
}

// ---------------------------------------------------------------------------
extern "C" void kernel_launch(void* const* d_in, const int* in_sizes, int n_in,
                              void* d_out, int out_size, void* d_ws, size_t ws_size,
                              hipStream_t stream) {
    const float* x      = (const float*)d_in[0];   // (B,T,D) fp32
    const float* levels = (const float*)d_in[1];   // (G,DEPTH,2,GD) fp32
    float* out  = (float*)d_out;                   // [x_q_ste | loss | indices]
    float* gram = (float*)d_ws;                    // 8 KB Gram workspace

    tpq_setup<<<GG, 256, 0, stream>>>(levels, gram, out);
    tpq_main<<<NTILES, 256, 0, stream>>>(x, levels, gram, out);
}